// HeteroGraphSAGE_68521908240493
// MI455X (gfx1250) — compile-verified
//
#include <hip/hip_runtime.h>
#include <math.h>

#define HDIM   128
#define NU     200000
#define NI     200000
#define NE     1000000
#define INU    256
#define INI    128
#define LN_EPS 1e-5f

typedef __attribute__((ext_vector_type(2))) float v2f;
typedef __attribute__((ext_vector_type(8))) float v8f;

// ---------------------------------------------------------------- fill zero
__global__ void fill_zero(float* __restrict__ p, int n) {
    int i = blockIdx.x * blockDim.x + threadIdx.x;
    if (i < n) p[i] = 0.0f;
}

// ---------------------------------------------------------------- edge scatter
// One wave (32 lanes) per edge; each lane handles 4 consecutive features.
// agg[dst] += xsrc[src]; cnt[dst] += 1
__global__ void __launch_bounds__(256)
edge_scatter(const float* __restrict__ xsrc,
             const int*   __restrict__ src,
             const int*   __restrict__ dst,
             float*       __restrict__ agg,
             float*       __restrict__ cnt,
             int nedges)
{
    int t    = blockIdx.x * blockDim.x + threadIdx.x;
    int e    = t >> 5;
    int lane = t & 31;
    if (e >= nedges) return;
    int s = src[e];
    int d = dst[e];
    const float4 v = ((const float4*)(xsrc + (long)s * HDIM))[lane];
    float* out = agg + (long)d * HDIM + lane * 4;
    atomicAdd(out + 0, v.x);
    atomicAdd(out + 1, v.y);
    atomicAdd(out + 2, v.z);
    atomicAdd(out + 3, v.w);
    if (lane == 0) atomicAdd(cnt + d, 1.0f);
}

// ---------------------------------------------------------------- WMMA GEMM
// Y[M x 128] = (rowscale . A) @ W1 + bias  (+ X2 @ W2)
//   A : M x K1 (K1 in {128,256}), staged through LDS (scale fused)
//   X2: M x 128 (optional), W1: K1 x 128, W2: 128 x 128
// grid.x = M/16 (M % 16 == 0), block = 256 threads = 8 waves.
// Each wave owns one 16-wide column strip; fp32 WMMA 16x16x4.
// Y may alias A (each block reads only its own 16 rows, before the barrier).
__global__ void __launch_bounds__(256)
sage_gemm(const float* __restrict__ A, int K1,
          const float* __restrict__ cnt,   // null -> scale 1
          const float* __restrict__ W1,
          const float* __restrict__ bias,
          const float* __restrict__ X2,    // null -> skip second GEMM
          const float* __restrict__ W2,
          float*       __restrict__ Y)
{
    __shared__ float ldsA[16 * 256];
    __shared__ float ldsB[16 * HDIM];

    const int  tid  = threadIdx.x;
    const int  lane = tid & 31;
    const int  wave = tid >> 5;
    const long row0 = (long)blockIdx.x * 16;

    // stage A tile, fusing the segment-mean divide
    {
        const int cpr = K1 >> 2;            // float4 per row
        const int nf4 = 16 * cpr;           // multiple of 256 -> uniform loop
        for (int idx = tid; idx < nf4; idx += 256) {
            int r  = idx / cpr;
            int c4 = idx - r * cpr;
            float4 v = ((const float4*)(A + (row0 + r) * (long)K1))[c4];
            if (cnt) {
                float sc = 1.0f / fmaxf(cnt[row0 + r], 1.0f);
                v.x *= sc; v.y *= sc; v.z *= sc; v.w *= sc;
            }
            ((float4*)(ldsA + r * K1))[c4] = v;
        }
    }
    if (X2) {
        for (int idx = tid; idx < 16 * (HDIM >> 2); idx += 256) {
            int r = idx >> 5, c4 = idx & 31;
            ((float4*)(ldsB + r * HDIM))[c4] =
                ((const float4*)(X2 + (row0 + r) * (long)HDIM))[c4];
        }
    }
    __syncthreads();

    const int m    = lane & 15;           // row within tile (A) / col (B,D)
    const int koff = (lane >> 4) << 1;    // 0 for lanes 0-15, 2 for 16-31
    const int n    = (wave << 4) + m;     // output column

    const float bv = bias[n];
    v8f acc = {bv, bv, bv, bv, bv, bv, bv, bv};

    for (int k = 0; k < K1; k += 4) {
        v2f a, b;
        a.x = ldsA[m * K1 + k + koff];
        a.y = ldsA[m * K1 + k + koff + 1];
        b.x = W1[(k + koff)     * HDIM + n];
        b.y = W1[(k + koff + 1) * HDIM + n];
        acc = __builtin_amdgcn_wmma_f32_16x16x4_f32(false, a, false, b,
                                                    (short)0, acc, false, false);
    }
    if (X2) {
        for (int k = 0; k < HDIM; k += 4) {
            v2f a, b;
            a.x = ldsB[m * HDIM + k + koff];
            a.y = ldsB[m * HDIM + k + koff + 1];
            b.x = W2[(k + koff)     * HDIM + n];
            b.y = W2[(k + koff + 1) * HDIM + n];
            acc = __builtin_amdgcn_wmma_f32_16x16x4_f32(false, a, false, b,
                                                        (short)0, acc, false, false);
        }
    }

    const long rbase = row0 + ((lane >> 4) << 3);
#pragma unroll
    for (int r = 0; r < 8; ++r) {
        Y[(rbase + r) * HDIM + n] = acc[r];
    }
}

// ---------------------------------------------------------------- LN+ReLU+res
// One wave per row: h[row] = relu(layernorm(t[row])*g + b) + h[row]
__global__ void __launch_bounds__(256)
ln_relu_res(const float* __restrict__ T,
            const float* __restrict__ g,
            const float* __restrict__ b,
            float*       __restrict__ Hbuf,
            int M)
{
    int gid  = blockIdx.x * blockDim.x + threadIdx.x;
    int row  = gid >> 5;
    int lane = threadIdx.x & 31;
    if (row >= M) return;

    const float4 t = ((const float4*)(T + (long)row * HDIM))[lane];
    float s  = t.x + t.y + t.z + t.w;
    float s2 = t.x * t.x + t.y * t.y + t.z * t.z + t.w * t.w;
#pragma unroll
    for (int off = 16; off > 0; off >>= 1) {
        s  += __shfl_xor(s,  off, 32);
        s2 += __shfl_xor(s2, off, 32);
    }
    const float mu  = s * (1.0f / HDIM);
    const float var = s2 * (1.0f / HDIM) - mu * mu;
    const float rs  = rsqrtf(var + LN_EPS);

    const float4 gg = ((const float4*)g)[lane];
    const float4 bb = ((const float4*)b)[lane];
    float4 h = ((const float4*)(Hbuf + (long)row * HDIM))[lane];
    float4 o;
    o.x = fmaxf((t.x - mu) * rs * gg.x + bb.x, 0.0f) + h.x;
    o.y = fmaxf((t.y - mu) * rs * gg.y + bb.y, 0.0f) + h.y;
    o.z = fmaxf((t.z - mu) * rs * gg.z + bb.z, 0.0f) + h.z;
    o.w = fmaxf((t.w - mu) * rs * gg.w + bb.w, 0.0f) + h.w;
    ((float4*)(Hbuf + (long)row * HDIM))[lane] = o;
}

// ---------------------------------------------------------------- launch
struct LayerPtrs {
    const float *ui_Wl, *ui_bl, *ui_Wr;
    const float *iu_Wl, *iu_bl, *iu_Wr;
    const float *lnu_g, *lnu_b, *lni_g, *lni_b;
};

extern "C" void kernel_launch(void* const* d_in, const int* in_sizes, int n_in,
                              void* d_out, int out_size, void* d_ws, size_t ws_size,
                              hipStream_t stream)
{
    (void)n_in; (void)out_size; (void)ws_size;

    const float *x_user, *x_item, *pwu, *pbu, *pwi, *pbi;
    const int   *ei_ui,  *ei_iu;
    LayerPtrs L[2];

    // Disambiguate harness flattening order from in_sizes:
    //   insertion order  -> in_sizes[0] == N_USER*IN_USER (51.2M)
    //   jax alphabetical -> in_sizes[0] == 2*E            (2M)
    if (in_sizes[0] == NU * INU) {
        x_user = (const float*)d_in[0];
        x_item = (const float*)d_in[1];
        ei_ui  = (const int*)  d_in[2];
        ei_iu  = (const int*)  d_in[3];
        pwu = (const float*)d_in[4]; pbu = (const float*)d_in[5];
        pwi = (const float*)d_in[6]; pbi = (const float*)d_in[7];
        for (int l = 0; l < 2; ++l) {
            int k = 8 + 10 * l;
            L[l].ui_Wl = (const float*)d_in[k + 0];
            L[l].ui_bl = (const float*)d_in[k + 1];
            L[l].ui_Wr = (const float*)d_in[k + 2];
            L[l].iu_Wl = (const float*)d_in[k + 3];
            L[l].iu_bl = (const float*)d_in[k + 4];
            L[l].iu_Wr = (const float*)d_in[k + 5];
            L[l].lnu_g = (const float*)d_in[k + 6];
            L[l].lnu_b = (const float*)d_in[k + 7];
            L[l].lni_g = (const float*)d_in[k + 8];
            L[l].lni_b = (const float*)d_in[k + 9];
        }
    } else {
        ei_iu = (const int*)d_in[0];
        ei_ui = (const int*)d_in[1];
        for (int l = 0; l < 2; ++l) {
            int k = 2 + 10 * l;
            L[l].iu_Wl = (const float*)d_in[k + 0];
            L[l].iu_Wr = (const float*)d_in[k + 1];
            L[l].iu_bl = (const float*)d_in[k + 2];
            L[l].lni_b = (const float*)d_in[k + 3];
            L[l].lni_g = (const float*)d_in[k + 4];
            L[l].lnu_b = (const float*)d_in[k + 5];
            L[l].lnu_g = (const float*)d_in[k + 6];
            L[l].ui_Wl = (const float*)d_in[k + 7];
            L[l].ui_Wr = (const float*)d_in[k + 8];
            L[l].ui_bl = (const float*)d_in[k + 9];
        }
        pwi = (const float*)d_in[22]; pbi = (const float*)d_in[23];
        pwu = (const float*)d_in[24]; pbu = (const float*)d_in[25];
        x_item = (const float*)d_in[26];
        x_user = (const float*)d_in[27];
    }

    // d_out holds h_user then h_item
    float* h_u = (float*)d_out;
    float* h_i = h_u + (long)NU * HDIM;

    // workspace: A_u (NU*128) | A_i (NI*128) | cnt (max(NU,NI))
    float* A_u = (float*)d_ws;
    float* A_i = A_u + (long)NU * HDIM;
    float* cnt = A_i + (long)NI * HDIM;

    const dim3 blk(256);
    const int nbig   = NU * HDIM;                 // 25.6M
    const int gz_big = (nbig + 255) / 256;
    const int gz_cnt = (NU + 255) / 256;
    const int g_edge = (NE * 32) / 256;
    const int g_ln   = (NU * 32) / 256;

    // input projections (straight into d_out)
    sage_gemm<<<NU / 16, blk, 0, stream>>>(x_user, INU, nullptr, pwu, pbu,
                                           nullptr, nullptr, h_u);
    sage_gemm<<<NI / 16, blk, 0, stream>>>(x_item, INI, nullptr, pwi, pbi,
                                           nullptr, nullptr, h_i);

    for (int l = 0; l < 2; ++l) {
        // users <- mean over incoming item edges (ei_iu: row0 src item, row1 dst user)
        fill_zero<<<gz_big, blk, 0, stream>>>(A_u, nbig);
        fill_zero<<<gz_cnt, blk, 0, stream>>>(cnt, NU);
        edge_scatter<<<g_edge, blk, 0, stream>>>(h_i, ei_iu, ei_iu + NE, A_u, cnt, NE);
        sage_gemm<<<NU / 16, blk, 0, stream>>>(A_u, HDIM, cnt, L[l].iu_Wl, L[l].iu_bl,
                                               h_u, L[l].iu_Wr, A_u); // in-place

        // items <- mean over incoming user edges (ei_ui: row0 src user, row1 dst item)
        fill_zero<<<gz_big, blk, 0, stream>>>(A_i, nbig);
        fill_zero<<<gz_cnt, blk, 0, stream>>>(cnt, NI);
        edge_scatter<<<g_edge, blk, 0, stream>>>(h_u, ei_ui, ei_ui + NE, A_i, cnt, NE);
        sage_gemm<<<NI / 16, blk, 0, stream>>>(A_i, HDIM, cnt, L[l].ui_Wl, L[l].ui_bl,
                                               h_i, L[l].ui_Wr, A_i); // in-place

        // norm -> relu -> residual, in-place on d_out
        ln_relu_res<<<g_ln, blk, 0, stream>>>(A_u, L[l].lnu_g, L[l].lnu_b, h_u, NU);
        ln_relu_res<<<g_ln, blk, 0, stream>>>(A_i, L[l].lni_g, L[l].lni_b, h_i, NI);
    }
}